// CRF_68539088110193
// MI455X (gfx1250) — compile-verified
//
#include <hip/hip_runtime.h>
#include <hip/hip_bf16.h>

typedef __attribute__((ext_vector_type(2))) float v2f;
typedef __attribute__((ext_vector_type(8))) float v8f;
typedef __attribute__((ext_vector_type(4))) int   v4i;

static constexpr int NB = 64;     // batch
static constexpr int NT = 512;    // time
static constexpr int ND = 2048;   // input dim
static constexpr int NF = 128;    // tags
static constexpr int NBT = NB * NT;

#define GLOBAL_AS __attribute__((address_space(1)))
#define LDS_AS    __attribute__((address_space(3)))

// --- async global->LDS staging (gfx1250 GLOBAL_LOAD_ASYNC_TO_LDS_B128) ------
#if __has_builtin(__builtin_amdgcn_global_load_async_to_lds_b128)
#define HAVE_ASYNC_LDS 1
#else
#define HAVE_ASYNC_LDS 0
#endif

static __device__ __forceinline__ void async_copy_b128(const float* gsrc, float* ldst)
{
#if HAVE_ASYNC_LDS
    GLOBAL_AS v4i* g = (GLOBAL_AS v4i*)(GLOBAL_AS float*)(float*)gsrc;
    LDS_AS    v4i* l = (LDS_AS v4i*)(LDS_AS float*)ldst;
    __builtin_amdgcn_global_load_async_to_lds_b128(g, l, 0, 0);
#else
    *(float4*)ldst = *(const float4*)gsrc;      // sync fallback
#endif
}

static __device__ __forceinline__ void wait_async_all()
{
#if HAVE_ASYNC_LDS
#if __has_builtin(__builtin_amdgcn_s_wait_asynccnt)
    __builtin_amdgcn_s_wait_asynccnt(0);
#else
    asm volatile("s_wait_asynccnt 0x0" ::: "memory");
#endif
#endif
}

// ---------------------------------------------------------------------------
// Kernel A: energy = X @ kernel + bias (+ boundary rows), via f32 WMMA.
// Block: 256 threads (8 wave32s). Block tile: M=16 rows x N=128 cols.
// Wave w owns N-tile [16w, 16w+16). K double-buffered through LDS in chunks
// of 64 using async global->LDS copies; two WMMA accumulator chains for ILP.
// ---------------------------------------------------------------------------
__global__ __launch_bounds__(256) void crf_energy_gemm(
    const float* __restrict__ X, const float* __restrict__ W,
    const float* __restrict__ bias, const float* __restrict__ lb,
    const float* __restrict__ rb, float* __restrict__ energy)
{
    __shared__ float Xs[2][16][68];    // stride 68: conflict-free A-frag reads
    __shared__ float Ws[2][64][136];   // stride 136: half-wave banks disjoint

    const int tid   = threadIdx.x;
    const int wave  = tid >> 5;
    const int lane  = tid & 31;
    const int mbase = blockIdx.x * 16;
    const int nbase = wave * 16;

    auto stage = [&](int buf, int k0) {
        // X tile 16x64: one b128 per thread
        {
            const int row = tid >> 4;
            const int c4  = (tid & 15) * 4;
            async_copy_b128(X + (size_t)(mbase + row) * ND + k0 + c4,
                            &Xs[buf][row][c4]);
        }
        // kernel tile 64x128: 8 b128 per thread, coalesced
        #pragma unroll
        for (int i = 0; i < 8; ++i) {
            const int f4  = tid + i * 256;
            const int row = f4 >> 5;
            const int col = (f4 & 31) * 4;
            async_copy_b128(W + (size_t)(k0 + row) * NF + col,
                            &Ws[buf][row][col]);
        }
    };

    stage(0, 0);
    wait_async_all();
    __syncthreads();

    // ISA 32-bit 16x4 A layout: lanes 0-15 -> K = kk,kk+1 ; lanes 16-31 -> kk+2,kk+3
    const int arow = lane & 15;
    const int koff = (lane >> 4) << 1;
    const int bcol = nbase + (lane & 15);

    v8f acc0 = {};
    v8f acc1 = {};
    for (int k0 = 0; k0 < ND; k0 += 64) {
        const int buf = (k0 >> 6) & 1;
        if (k0 + 64 < ND) stage(buf ^ 1, k0 + 64);   // prefetch next chunk

        #pragma unroll
        for (int kk = 0; kk < 64; kk += 8) {
            v2f a, b;
            a.x = Xs[buf][arow][kk + koff];
            a.y = Xs[buf][arow][kk + koff + 1];
            b.x = Ws[buf][kk + koff][bcol];
            b.y = Ws[buf][kk + koff + 1][bcol];
            acc0 = __builtin_amdgcn_wmma_f32_16x16x4_f32(
                false, a, false, b, (short)0, acc0, false, false);

            a.x = Xs[buf][arow][kk + 4 + koff];
            a.y = Xs[buf][arow][kk + 4 + koff + 1];
            b.x = Ws[buf][kk + 4 + koff][bcol];
            b.y = Ws[buf][kk + 4 + koff + 1][bcol];
            acc1 = __builtin_amdgcn_wmma_f32_16x16x4_f32(
                false, a, false, b, (short)0, acc1, false, false);
        }

        wait_async_all();
        __syncthreads();
    }
    const v8f acc = acc0 + acc1;

    // C/D layout: VGPR r: lanes 0-15 -> M=r, lanes 16-31 -> M=r+8 ; N = lane&15
    const int n  = nbase + (lane & 15);
    const int mo = (lane >> 4) << 3;
    const float bn = bias[n];
    const float ln = lb[n];
    const float rn = rb[n];
    #pragma unroll
    for (int r = 0; r < 8; ++r) {
        const int gm = mbase + r + mo;
        const int t  = gm & (NT - 1);           // tiles never straddle a batch
        float v = acc[r] + bn;
        if (t == 0)      v += ln;
        if (t == NT - 1) v += rn;
        energy[(size_t)gm * NF + n] = v;
    }
}

// ---------------------------------------------------------------------------
// Kernel B: forward/backward logsumexp scan.
//   new[f] = m + log( sum_j exp(prev[j]-e[j]-m) * exp(-chain[j][f]) )
// E = exp(-chain) cached in LDS (64 KB). One 128-thread block per (dir,batch).
// Parity double-buffered LDS -> 2 barriers/step. Energies software-pipelined.
// ---------------------------------------------------------------------------
__global__ __launch_bounds__(128) void crf_scan(
    const float* __restrict__ energy, const float* __restrict__ chain,
    float* __restrict__ alpha, float* __restrict__ beta)
{
    __shared__ float E[NF * NF];
    __shared__ float wbuf[2][NF];
    __shared__ float warr[2][4];

    const int tid  = threadIdx.x;
    const int lane = tid & 31;
    const int wave = tid >> 5;
    const int b    = blockIdx.x & (NB - 1);
    const int dir  = blockIdx.x >> 6;

    const float* eb = energy + (size_t)b * NT * NF;
    float* ob = (dir ? beta : alpha) + (size_t)b * NT * NF;

    for (int i = tid; i < NF * NF; i += 128)
        E[i] = __expf(-chain[i]);
    __syncthreads();

    float prev  = 0.0f;
    float e_cur = eb[(size_t)(dir ? NT - 1 : 0) * NF + tid];

    for (int s = 0; s < NT; ++s) {
        const int idx = dir ? (NT - 1 - s) : s;
        // software pipeline the next step's energy load; prefetch 8 ahead
        float e_nxt = 0.0f;
        if (s + 1 < NT) {
            const int ni = dir ? (idx - 1) : (idx + 1);
            e_nxt = eb[(size_t)ni * NF + tid];
        }
        if (s + 8 < NT) {
            const int pf = dir ? (idx - 8) : (idx + 8);
            __builtin_prefetch(eb + (size_t)pf * NF + tid, 0, 3);
        }

        const int par = s & 1;
        const float pm = prev - e_cur;

        // block max of pm: wave32 shfl reduce, then 4-wide LDS combine
        float wm = pm;
        #pragma unroll
        for (int off = 16; off >= 1; off >>= 1)
            wm = fmaxf(wm, __shfl_xor(wm, off, 32));
        if (lane == 0) warr[par][wave] = wm;
        __syncthreads();
        const float m = fmaxf(fmaxf(warr[par][0], warr[par][1]),
                              fmaxf(warr[par][2], warr[par][3]));

        wbuf[par][tid] = __expf(pm - m);
        __syncthreads();

        float acc = 0.0f;
        #pragma unroll 16
        for (int j = 0; j < NF; ++j)
            acc = fmaf(wbuf[par][j], E[j * NF + tid], acc);

        prev = m + __logf(acc);
        ob[(size_t)idx * NF + tid] = prev;
        e_cur = e_nxt;
    }
}

// ---------------------------------------------------------------------------
// Kernel C: margin = -(alpha_sr + energy + beta_sl); softmax over F.
// One 128-thread block per (b,t) row.
// ---------------------------------------------------------------------------
__global__ __launch_bounds__(128) void crf_margin_softmax(
    const float* __restrict__ energy, const float* __restrict__ alpha,
    const float* __restrict__ beta, float* __restrict__ out)
{
    __shared__ float rmax[4];
    __shared__ float rsum[4];

    const int tid  = threadIdx.x;
    const int lane = tid & 31;
    const int wave = tid >> 5;
    const int row  = blockIdx.x;          // row = b*NT + t
    const int t    = row & (NT - 1);
    const size_t base = (size_t)row * NF + tid;

    float v = energy[base];
    if (t > 0)      v += alpha[base - NF];   // alpha shifted right
    if (t < NT - 1) v += beta[base + NF];    // beta shifted left
    float mg = -v;

    float m = mg;
    #pragma unroll
    for (int off = 16; off >= 1; off >>= 1)
        m = fmaxf(m, __shfl_xor(m, off, 32));
    if (lane == 0) rmax[wave] = m;
    __syncthreads();
    m = fmaxf(fmaxf(rmax[0], rmax[1]), fmaxf(rmax[2], rmax[3]));

    const float e = __expf(mg - m);
    float s = e;
    #pragma unroll
    for (int off = 16; off >= 1; off >>= 1)
        s += __shfl_xor(s, off, 32);
    if (lane == 0) rsum[wave] = s;
    __syncthreads();
    s = rsum[0] + rsum[1] + rsum[2] + rsum[3];

    out[base] = e / s;
}

// ---------------------------------------------------------------------------
extern "C" void kernel_launch(void* const* d_in, const int* in_sizes, int n_in,
                              void* d_out, int out_size, void* d_ws, size_t ws_size,
                              hipStream_t stream)
{
    const float* X     = (const float*)d_in[0];
    const float* W     = (const float*)d_in[1];
    const float* chain = (const float*)d_in[2];
    const float* bias  = (const float*)d_in[3];
    const float* lb    = (const float*)d_in[4];
    const float* rb    = (const float*)d_in[5];
    float* out = (float*)d_out;

    float* energy = (float*)d_ws;                       // 16 MB
    float* alpha  = energy + (size_t)NBT * NF;          // 16 MB
    float* beta   = alpha  + (size_t)NBT * NF;          // 16 MB

    crf_energy_gemm<<<NBT / 16, 256, 0, stream>>>(X, W, bias, lb, rb, energy);
    crf_scan<<<2 * NB, 128, 0, stream>>>(energy, chain, alpha, beta);
    crf_margin_softmax<<<NBT, 128, 0, stream>>>(energy, alpha, beta, out);
}